// NNConvNet_90400471646334
// MI455X (gfx1250) — compile-verified
//
#include <hip/hip_runtime.h>

typedef float v2f __attribute__((ext_vector_type(2)));
typedef float v8f __attribute__((ext_vector_type(8)));

// D = A(16x4 f32) * B(4x16 f32) + C(16x16 f32), full-precision WMMA on gfx1250.
__device__ __forceinline__ v8f wmma4(v2f a, v2f b, v8f c) {
  return __builtin_amdgcn_wmma_f32_16x16x4_f32(false, a, false, b, (short)0, c,
                                               false, false);
}

__device__ __forceinline__ float lrelu(float v) { return fmaxf(v, 0.01f * v); }

// gfx1250 f32 WMMA operand layouts (CDNA5 ISA 7.12.2):
//  A 16x4 : VGPR0 lanes0-15 = (M=lane,K=0), lanes16-31 = (M=lane-16,K=2); VGPR1 = K=1/K=3
//  B 4x16 : VGPR0 lanes0-15 = (K=0,N=lane), lanes16-31 = (K=2,N=lane-16); VGPR1 = K=1/K=3
//  C/D 16x16: VGPR r, lanes0-15 = (M=r, N=lane), lanes16-31 = (M=r+8, N=lane-16)
//  -> bias (per-column) is a splat across the 8 C VGPRs; per-element C-init is legal.

// ---------------- Stage 1: h3 = lrelu(lrelu(x@W_in+b_in)@W_enc+b_enc); zero agg ----
__global__ void __launch_bounds__(32)
nnconv_stage1(const float* __restrict__ x,
              const float* __restrict__ W_in, const float* __restrict__ b_in,
              const float* __restrict__ W_enc, const float* __restrict__ b_enc,
              float* __restrict__ h3, float* __restrict__ agg, int N) {
  __shared__ __align__(16) float hs[16][32];
  const int lane = threadIdx.x;
  const int n0   = blockIdx.x * 16;
  const int m    = lane & 15;
  const int hsel = lane >> 4;      // lane half
  const int khi  = hsel << 1;      // K offset within a 4-wide K step
  const int mload = min(n0 + m, N - 1);

  // GEMM1: [16x64] @ [64x32]; bias folded into C-init
  const float bi0 = b_in[m], bi1 = b_in[16 + m];
  v8f c0, c1;
#pragma unroll
  for (int r = 0; r < 8; ++r) { c0[r] = bi0; c1[r] = bi1; }
  for (int kb = 0; kb < 64; kb += 4) {
    const int k = kb + khi;
    v2f a = *(const v2f*)(x + (size_t)mload * 64 + k);
    v2f b0, b1;
    b0.x = W_in[k * 32 + m];       b0.y = W_in[(k + 1) * 32 + m];
    b1.x = W_in[k * 32 + 16 + m];  b1.y = W_in[(k + 1) * 32 + 16 + m];
    c0 = wmma4(a, b0, c0);
    c1 = wmma4(a, b1, c1);
  }
  // lrelu, transpose C-layout -> [node][ch] in LDS for next GEMM's A
#pragma unroll
  for (int r = 0; r < 8; ++r) {
    const int row = r + (hsel << 3);
    hs[row][m]      = lrelu(c0[r]);
    hs[row][16 + m] = lrelu(c1[r]);
  }
  __syncthreads();

  // GEMM2: [16x32] @ [32x32]; bias folded into C-init
  const float be0 = b_enc[m], be1 = b_enc[16 + m];
  v8f d0, d1;
#pragma unroll
  for (int r = 0; r < 8; ++r) { d0[r] = be0; d1[r] = be1; }
  for (int kb = 0; kb < 32; kb += 4) {
    const int k = kb + khi;
    v2f a; a.x = hs[m][k]; a.y = hs[m][k + 1];
    v2f b0, b1;
    b0.x = W_enc[k * 32 + m];       b0.y = W_enc[(k + 1) * 32 + m];
    b1.x = W_enc[k * 32 + 16 + m];  b1.y = W_enc[(k + 1) * 32 + 16 + m];
    d0 = wmma4(a, b0, d0);
    d1 = wmma4(a, b1, d1);
  }
#pragma unroll
  for (int r = 0; r < 8; ++r) {
    const int row = r + (hsel << 3);
    if (n0 + row < N) {
      h3[(size_t)(n0 + row) * 32 + m]      = lrelu(d0[r]);
      h3[(size_t)(n0 + row) * 32 + 16 + m] = lrelu(d1[r]);
    }
  }
  // zero this tile's slice of agg (fresh every launch -> deterministic)
  for (int j = lane; j < 512; j += 32) {
    const size_t idx = (size_t)n0 * 32 + j;
    if (idx < (size_t)N * 32) agg[idx] = 0.f;
  }
}

// ---------------- Stage 2 (edges): fused w=lrelu(ea@W_edge+b); msg=h3[src]@w; scatter ----
// 4 edge tiles (64 edges) per wave so the W_edge B-operand + bias loads for each
// channel i are amortized 4x; bias enters through the WMMA C operand.
#define ETILES 4
__global__ void __launch_bounds__(32)
nnconv_edge(const float* __restrict__ h3,
            const long long* __restrict__ ei,     // [2,E] int64
            const float* __restrict__ edge_attr,  // [E,8]
            const float* __restrict__ W_edge,     // [8,1024]
            const float* __restrict__ b_edge,     // [1024]
            float* __restrict__ agg, int E) {
  __shared__ __align__(16) float hsrc[64][32];
  __shared__ long long sidx[64];
  const int lane = threadIdx.x;
  const int e0   = blockIdx.x * (ETILES * 16);
  const int m    = lane & 15;
  const int hsel = lane >> 4;
  const int khi  = hsel << 1;
  const long long* __restrict__ srcp = ei;
  const long long* __restrict__ dstp = ei + E;

  // stage src indices once
  for (int j = lane; j < 64; j += 32) sidx[j] = srcp[min(e0 + j, E - 1)];
  __syncthreads();

  // A operands: 4 tiles of edge_attr (16 edges x 8), two K-steps of 4 each
  v2f a0[ETILES], a1[ETILES];
#pragma unroll
  for (int t = 0; t < ETILES; ++t) {
    const int eload = min(e0 + t * 16 + m, E - 1);
    a0[t] = *(const v2f*)(edge_attr + (size_t)eload * 8 + khi);
    a1[t] = *(const v2f*)(edge_attr + (size_t)eload * 8 + 4 + khi);
  }

  // gather h3[src] rows (64 x 32 f32) into LDS, float4-vectorized
  for (int q = lane; q < 512; q += 32) {
    const int row = q >> 3, p = q & 7;
    *(float4*)(&hsrc[row][p * 4]) =
        *(const float4*)(h3 + (size_t)sidx[row] * 32 + p * 4);
  }
  __syncthreads();

  v8f acc0[ETILES], acc1[ETILES];   // msg accumulators, o-halves [0,16) / [16,32)
#pragma unroll
  for (int t = 0; t < ETILES; ++t)
#pragma unroll
    for (int r = 0; r < 8; ++r) { acc0[t][r] = 0.f; acc1[t][r] = 0.f; }

  for (int i = 0; i < 32; ++i) {
    const int c0b = i * 32, c1b = i * 32 + 16;
    // B operand for channel i: shared by all 4 edge tiles (L1/WGP$-resident 32KB)
    v2f b00, b01, b10, b11;
    b00.x = W_edge[khi * 1024 + c0b + m];        b00.y = W_edge[(khi + 1) * 1024 + c0b + m];
    b01.x = W_edge[khi * 1024 + c1b + m];        b01.y = W_edge[(khi + 1) * 1024 + c1b + m];
    b10.x = W_edge[(4 + khi) * 1024 + c0b + m];  b10.y = W_edge[(5 + khi) * 1024 + c0b + m];
    b11.x = W_edge[(4 + khi) * 1024 + c1b + m];  b11.y = W_edge[(5 + khi) * 1024 + c1b + m];
    const float bb0 = b_edge[c0b + m];
    const float bb1 = b_edge[c1b + m];
#pragma unroll
    for (int t = 0; t < ETILES; ++t) {
      v8f w0, w1;
#pragma unroll
      for (int r = 0; r < 8; ++r) { w0[r] = bb0; w1[r] = bb1; }  // bias as C-init
      w0 = wmma4(a0[t], b00, w0);
      w1 = wmma4(a0[t], b01, w1);
      w0 = wmma4(a1[t], b10, w0);
      w1 = wmma4(a1[t], b11, w1);
#pragma unroll
      for (int r = 0; r < 8; ++r) {
        const int row = t * 16 + r + (hsel << 3);  // edge row this lane holds
        const float hv = hsrc[row][i];             // LDS broadcast per half-wave
        acc0[t][r] = fmaf(hv, lrelu(w0[r]), acc0[t][r]);
        acc1[t][r] = fmaf(hv, lrelu(w1[r]), acc1[t][r]);
      }
    }
  }
  // scatter-add messages into agg[dst]
#pragma unroll
  for (int t = 0; t < ETILES; ++t) {
#pragma unroll
    for (int r = 0; r < 8; ++r) {
      const int row = t * 16 + r + (hsel << 3);
      if (e0 + row < E) {
        const long long d = dstp[e0 + row];
        atomicAdd(&agg[(size_t)d * 32 + m],      acc0[t][r]);
        atomicAdd(&agg[(size_t)d * 32 + 16 + m], acc1[t][r]);
      }
    }
  }
}

// ---------------- Stage 3: agg + h3@W_root + b; lrelu; @W_dec + b; @W_out; log_softmax ----
__global__ void __launch_bounds__(32)
nnconv_stage2(const float* __restrict__ h3, const float* __restrict__ agg,
              const float* __restrict__ W_root, const float* __restrict__ b_conv,
              const float* __restrict__ W_dec, const float* __restrict__ b_dec,
              const float* __restrict__ W_out, const float* __restrict__ b_out,
              float* __restrict__ out, int N) {
  __shared__ __align__(16) float hs[16][32];
  __shared__ __align__(16) float h5[16][32];
  const int lane = threadIdx.x;
  const int n0   = blockIdx.x * 16;
  const int m    = lane & 15;
  const int hsel = lane >> 4;
  const int khi  = hsel << 1;

  for (int q = lane; q < 64; q += 32) {
    const int e = q >> 2, p = q & 3;
    const int n = min(n0 + e, N - 1);
    *(float4*)(&hs[e][p * 4]) = *(const float4*)(h3 + (size_t)n * 32 + p * 4);
  }
  __syncthreads();

  // root GEMM: C-init = agg + b_conv (per-element), then accumulate h3 @ W_root
  const float bc0 = b_conv[m], bc1 = b_conv[16 + m];
  v8f c0, c1;
#pragma unroll
  for (int r = 0; r < 8; ++r) {
    const int row = r + (hsel << 3);
    const int nn  = min(n0 + row, N - 1);
    c0[r] = agg[(size_t)nn * 32 + m]      + bc0;
    c1[r] = agg[(size_t)nn * 32 + 16 + m] + bc1;
  }
  for (int kb = 0; kb < 32; kb += 4) {
    const int k = kb + khi;
    v2f a; a.x = hs[m][k]; a.y = hs[m][k + 1];
    v2f b0, b1;
    b0.x = W_root[k * 32 + m];       b0.y = W_root[(k + 1) * 32 + m];
    b1.x = W_root[k * 32 + 16 + m];  b1.y = W_root[(k + 1) * 32 + 16 + m];
    c0 = wmma4(a, b0, c0);
    c1 = wmma4(a, b1, c1);
  }
  // h4 = lrelu(conv out); re-transpose into LDS
  float v0[8], v1[8];
#pragma unroll
  for (int r = 0; r < 8; ++r) { v0[r] = lrelu(c0[r]); v1[r] = lrelu(c1[r]); }
  __syncthreads();
#pragma unroll
  for (int r = 0; r < 8; ++r) {
    const int row = r + (hsel << 3);
    hs[row][m]      = v0[r];
    hs[row][16 + m] = v1[r];
  }
  __syncthreads();

  // decoder GEMM: h5 = h4 @ W_dec + b_dec (bias via C-init; no activation after)
  const float bd0 = b_dec[m], bd1 = b_dec[16 + m];
  v8f d0, d1;
#pragma unroll
  for (int r = 0; r < 8; ++r) { d0[r] = bd0; d1[r] = bd1; }
  for (int kb = 0; kb < 32; kb += 4) {
    const int k = kb + khi;
    v2f a; a.x = hs[m][k]; a.y = hs[m][k + 1];
    v2f b0, b1;
    b0.x = W_dec[k * 32 + m];       b0.y = W_dec[(k + 1) * 32 + m];
    b1.x = W_dec[k * 32 + 16 + m];  b1.y = W_dec[(k + 1) * 32 + 16 + m];
    d0 = wmma4(a, b0, d0);
    d1 = wmma4(a, b1, d1);
  }
#pragma unroll
  for (int r = 0; r < 8; ++r) {
    const int row = r + (hsel << 3);
    h5[row][m]      = d0[r];
    h5[row][16 + m] = d1[r];
  }
  __syncthreads();

  // lin_out + log_softmax([0, z]) ; lanes 0..15 each own one node
  if (lane < 16) {
    const int n = n0 + m;
    if (n < N) {
      float z = b_out[0];
#pragma unroll
      for (int c = 0; c < 32; ++c) z = fmaf(h5[m][c], W_out[c], z);
      const float mx  = fmaxf(z, 0.f);
      const float lse = mx + logf(expf(0.f - mx) + expf(z - mx));
      out[(size_t)n * 2 + 0] = 0.f - lse;
      out[(size_t)n * 2 + 1] = z   - lse;
    }
  }
}

extern "C" void kernel_launch(void* const* d_in, const int* in_sizes, int n_in,
                              void* d_out, int out_size, void* d_ws, size_t ws_size,
                              hipStream_t stream) {
  const float*     x         = (const float*)d_in[0];
  const long long* edge_index= (const long long*)d_in[1];   // int64 [2,E]
  const float*     edge_attr = (const float*)d_in[2];
  const float*     W_in      = (const float*)d_in[3];
  const float*     b_in      = (const float*)d_in[4];
  const float*     W_enc     = (const float*)d_in[5];
  const float*     b_enc     = (const float*)d_in[6];
  const float*     W_edge    = (const float*)d_in[7];
  const float*     b_edge    = (const float*)d_in[8];
  const float*     W_root    = (const float*)d_in[9];
  const float*     b_conv    = (const float*)d_in[10];
  const float*     W_dec     = (const float*)d_in[11];
  const float*     b_dec     = (const float*)d_in[12];
  const float*     W_out     = (const float*)d_in[13];
  const float*     b_out     = (const float*)d_in[14];
  float*           out       = (float*)d_out;

  const int N = in_sizes[0] / 64;   // x is [N,64]
  const int E = in_sizes[2] / 8;    // edge_attr is [E,8]

  float* h3  = (float*)d_ws;                 // [N,32]
  float* agg = h3 + (size_t)N * 32;          // [N,32]

  const int node_tiles = (N + 15) / 16;
  const int edge_blocks = (E + ETILES * 16 - 1) / (ETILES * 16);

  nnconv_stage1<<<node_tiles, 32, 0, stream>>>(x, W_in, b_in, W_enc, b_enc,
                                               h3, agg, N);
  nnconv_edge<<<edge_blocks, 32, 0, stream>>>(h3, edge_index, edge_attr,
                                              W_edge, b_edge, agg, E);
  nnconv_stage2<<<node_tiles, 32, 0, stream>>>(h3, agg, W_root, b_conv,
                                               W_dec, b_dec, W_out, b_out,
                                               out, N);
}